// Embedder_2284922602000
// MI455X (gfx1250) — compile-verified
//
#include <hip/hip_runtime.h>

// Embedder for MI455X (gfx1250, wave32).
// Numeric path = skinny GEMM (M=positions, K=16, N=64) on V_WMMA_F32_16X16X4_F32
// (f32 A/B/C -> exact f32 math, matches reference). Token path = embedding
// gather out of the L2-resident table (8 MB << 192 MB L2).
// Memory-bound: 134 MB of output stores dominate (~6 us floor @ 23.3 TB/s),
// so the output is written as fully-contiguous 512B global_store_b128 per wave
// (numeric results staged through LDS to undo the WMMA D-layout).

typedef __attribute__((ext_vector_type(2))) float v2f;
typedef __attribute__((ext_vector_type(8))) float v8f;

#define EMBED_DIM 64
#define HIDDEN    16
#define ROWSTRIDE 68   // floats; 272B row stride: 16B-aligned, conflict-free b32 scatter

__global__ void __launch_bounds__(256)
embedder_wmma_kernel(const float* __restrict__ ids,    // (B*S) token id or numeric value
                     const int*   __restrict__ mask,   // (B*S) 1=token, 0=numeric
                     const float* __restrict__ table,  // (VOCAB, 64)
                     const float* __restrict__ w1,     // (16,1)
                     const float* __restrict__ b1,     // (16,)
                     const float* __restrict__ w2,     // (64,16) row-major
                     const float* __restrict__ b2,     // (64,)
                     float*       __restrict__ out,    // (B*S, 64)
                     int n_pos, int n_tiles)
{
    __shared__ float smem[8 * 16 * ROWSTRIDE];         // 8 waves x 16 rows x 68 floats

    const int lane  = threadIdx.x & 31;
    const int wave  = threadIdx.x >> 5;
    const int wpb   = blockDim.x >> 5;
    const int gwave = blockIdx.x * wpb + wave;
    const int ngw   = gridDim.x * wpb;

    const int half = lane >> 4;     // 0: lanes 0-15, 1: lanes 16-31
    const int ln   = lane & 15;

    float* __restrict__ lrow = smem + wave * (16 * ROWSTRIDE);

    // ---- Preload B matrices for all 16 WMMAs (t = N-tile 0..3, j = K-slab 0..3).
    // B is 4x16 f32 in 2 VGPRs: VGPR0 holds K-row (4j + 2*half), VGPR1 the next,
    // column n = ln.  B[k][n] = w2[(16t+n)*16 + k]  (out = h @ w2^T).
    v2f Bm[4][4];
#pragma unroll
    for (int t = 0; t < 4; ++t) {
        const int d = t * 16 + ln;
#pragma unroll
        for (int j = 0; j < 4; ++j) {
            const int k0 = 4 * j + 2 * half;
            Bm[t][j].x = w2[d * HIDDEN + k0];
            Bm[t][j].y = w2[d * HIDDEN + k0 + 1];
        }
    }

    // Bias broadcast per N-tile: C[m][n] init = b2[16t + n] (same in all 8 C regs).
    float biasv[4];
#pragma unroll
    for (int t = 0; t < 4; ++t) biasv[t] = b2[t * 16 + ln];

    // First-layer weights this lane needs for A construction (k = 4j + 2*half + {0,1}).
    float w1v[4][2], b1v[4][2];
#pragma unroll
    for (int j = 0; j < 4; ++j) {
        const int k0 = 4 * j + 2 * half;
        w1v[j][0] = w1[k0];     b1v[j][0] = b1[k0];
        w1v[j][1] = w1[k0 + 1]; b1v[j][1] = b1[k0 + 1];
    }

    for (int tile = gwave; tile < n_tiles; tile += ngw) {
        const int  base = tile * 16;
        const bool full = (base + 16 <= n_pos);        // uniform per wave

        // x for position m = base + ln (replicated across both lane halves).
        int xp = base + ln;
        if (!full && xp >= n_pos) xp = n_pos - 1;      // clamp keeps EXEC all-ones
        const float x = ids[xp];

        // Accumulators start at bias.
        v8f acc[4];
#pragma unroll
        for (int t = 0; t < 4; ++t)
#pragma unroll
            for (int r = 0; r < 8; ++r) acc[t][r] = biasv[t];

        // 4 K-slabs x 4 N-tiles = 16 WMMAs. A rebuilt per slab from the scalar
        // MLP first layer: h[m][k] = relu(w1[k]*x[m] + b1[k]).
#pragma unroll
        for (int j = 0; j < 4; ++j) {
            float h0 = fmaf(w1v[j][0], x, b1v[j][0]); h0 = h0 > 0.f ? h0 : 0.f;
            float h1 = fmaf(w1v[j][1], x, b1v[j][1]); h1 = h1 > 0.f ? h1 : 0.f;
            v2f A; A.x = h0; A.y = h1;
#pragma unroll
            for (int t = 0; t < 4; ++t) {
                acc[t] = __builtin_amdgcn_wmma_f32_16x16x4_f32(
                    /*neg_a=*/false, A, /*neg_b=*/false, Bm[t][j],
                    /*c_mod=*/(short)0, acc[t],
                    /*reuse_a=*/false, /*reuse_b=*/false);
            }
        }

        if (full) {
            // ---- Stage numeric result in LDS (undo WMMA D layout).
            // D: reg r, lane -> row m = r + 8*half, col n = ln, dim = 16t + n.
            // Bank( m*68 + t*16 + ln ) is conflict-free across the wave.
#pragma unroll
            for (int t = 0; t < 4; ++t)
#pragma unroll
                for (int r = 0; r < 8; ++r) {
                    const int m = r + 8 * half;
                    lrow[m * ROWSTRIDE + t * 16 + ln] = acc[t][r];
                }
            asm volatile("" ::: "memory");  // keep DS store->load program order

            // ---- Select + 128b store: lanes 0-15 cover row q(=2i), lanes 16-31
            // row q+1 -> one contiguous 512B global_store_b128 per iteration.
#pragma unroll
            for (int i = 0; i < 8; ++i) {
                const int q = 2 * i + half;
                const int p = base + q;
                const int mk = mask[p];                 // uniform per half-wave
                float4 v;
                if (mk) {
                    const int id = (int)ids[p];         // row 0 is zeroed pad row
                    v = *(const float4*)(table + (size_t)id * EMBED_DIM + ln * 4);
                } else {
                    v = *(const float4*)(lrow + q * ROWSTRIDE + ln * 4);
                }
                *(float4*)(out + (size_t)p * EMBED_DIM + ln * 4) = v;
            }
        } else {
            // ---- Rare tail tile: guarded scalar stores straight from registers.
#pragma unroll
            for (int r = 0; r < 8; ++r) {
                const int p = base + r + 8 * half;
                if (p < n_pos) {
                    const int mk = mask[p];
                    float* __restrict__ orow = out + (size_t)p * EMBED_DIM;
                    if (mk) {
                        const int id = (int)ids[p];
                        const float* __restrict__ trow = table + (size_t)id * EMBED_DIM;
#pragma unroll
                        for (int t = 0; t < 4; ++t) orow[t * 16 + ln] = trow[t * 16 + ln];
                    } else {
#pragma unroll
                        for (int t = 0; t < 4; ++t) orow[t * 16 + ln] = acc[t][r];
                    }
                }
            }
        }
    }
}

extern "C" void kernel_launch(void* const* d_in, const int* in_sizes, int n_in,
                              void* d_out, int out_size, void* d_ws, size_t ws_size,
                              hipStream_t stream) {
    const float* ids   = (const float*)d_in[0];
    const int*   mask  = (const int*)  d_in[1];
    const float* table = (const float*)d_in[2];
    const float* w1    = (const float*)d_in[3];
    const float* b1    = (const float*)d_in[4];
    const float* w2    = (const float*)d_in[5];
    const float* b2    = (const float*)d_in[6];
    float* out = (float*)d_out;

    const int n_pos   = in_sizes[0];                 // B*S = 524288
    const int n_tiles = (n_pos + 15) / 16;           // 32768

    const int block = 256;                           // 8 wave32s
    int blocks = (n_tiles + 7) / 8;                  // one tile per wave
    if (blocks > 4096) blocks = 4096;
    if (blocks < 1) blocks = 1;

    embedder_wmma_kernel<<<blocks, block, 0, stream>>>(
        ids, mask, table, w1, b1, w2, b2, out, n_pos, n_tiles);
}